// NT_Xent2_62319975465315
// MI455X (gfx1250) — compile-verified
//
#include <hip/hip_runtime.h>
#include <math.h>

// NT-Xent2 for MI455X (gfx1250, wave32).
// Strategy: neg[n] = <x_hat[n], sum_m m_hat[m]>  (row-sum commutes through GEMM)
// -> one 64MB streaming pass over the memory bank instead of a 137-GFLOP GEMM.
// Per-pair reductions (pair dot, two squared norms, two dots-with-s) are computed
// on the matrix pipe via V_WMMA_F32_16X16X4_F32 Gram products (full fp32).

#define TEMPR 0.1f
#define BB    2048
#define DD    1024
#define MMR   16384
#define EPSR  1e-8f
#define NB1   256   // blocks in kernel 1 (partial-sum buffers)

typedef float v2f __attribute__((ext_vector_type(2)));
typedef float v8f __attribute__((ext_vector_type(8)));

// ---------------------------------------------------------------------------
// Kernel 1: partial sums of L2-normalized memory-bank rows.
// 256 blocks x 256 threads (8 waves). Each wave owns 8 rows; a wave processes
// one row at a time: lane l holds columns 4*(l+32*j), j=0..7 (coalesced float4).
// Squared norm via 5 shfl_xor steps (wave32), then scaled accumulate in regs.
// Waves combine through LDS with one barrier -> per-block partial vector.
// ---------------------------------------------------------------------------
__global__ __launch_bounds__(256) void k1_mem_partial(const float* __restrict__ mem,
                                                      float* __restrict__ partial) {
    const int lane = threadIdx.x & 31;
    const int w    = threadIdx.x >> 5;
    const int gw   = blockIdx.x * 8 + w;            // global wave id 0..2047

    float4 acc[8];
#pragma unroll
    for (int j = 0; j < 8; ++j) acc[j] = make_float4(0.f, 0.f, 0.f, 0.f);

    for (int rep = 0; rep < 8; ++rep) {
        const int row = gw + rep * 2048;            // 0..16383
        const float* rp = mem + (size_t)row * DD;
        float4 v[8];
        float ss = 0.f;
#pragma unroll
        for (int j = 0; j < 8; ++j) {
            v[j] = *(const float4*)(rp + 4 * (lane + 32 * j));
            ss += v[j].x * v[j].x + v[j].y * v[j].y + v[j].z * v[j].z + v[j].w * v[j].w;
        }
#pragma unroll
        for (int m = 16; m >= 1; m >>= 1) ss += __shfl_xor(ss, m, 32);
        const float scale = 1.0f / fmaxf(sqrtf(ss), EPSR);
#pragma unroll
        for (int j = 0; j < 8; ++j) {
            acc[j].x += v[j].x * scale;
            acc[j].y += v[j].y * scale;
            acc[j].z += v[j].z * scale;
            acc[j].w += v[j].w * scale;
        }
    }

    __shared__ float4 lds4[8][8][32];               // 32 KB
#pragma unroll
    for (int j = 0; j < 8; ++j) lds4[w][j][lane] = acc[j];
    __syncthreads();

    // 256 threads: thread t combines (j = t>>5, l = t&31) across the 8 waves.
    const int jj = threadIdx.x >> 5;
    const int ll = threadIdx.x & 31;
    float4 r = make_float4(0.f, 0.f, 0.f, 0.f);
#pragma unroll
    for (int ww = 0; ww < 8; ++ww) {
        float4 t = lds4[ww][jj][ll];
        r.x += t.x; r.y += t.y; r.z += t.z; r.w += t.w;
    }
    *(float4*)(partial + (size_t)blockIdx.x * DD + 4 * (ll + 32 * jj)) = r;
}

// ---------------------------------------------------------------------------
// Kernel 2: s[d] = sum over the 256 per-block partials (fixed order).
// ---------------------------------------------------------------------------
__global__ __launch_bounds__(256) void k2_sum_partials(const float* __restrict__ partial,
                                                       float* __restrict__ s) {
    const int d = blockIdx.x * 256 + threadIdx.x;
    float acc = 0.f;
    for (int b = 0; b < NB1; ++b) acc += partial[(size_t)b * DD + d];
    s[d] = acc;
}

// ---------------------------------------------------------------------------
// Kernel 3: one wave per 16 pairs. Five fp32 WMMA accumulators over K=1024:
//   Cp = X1*X2^T (diag = pair dots), C1 = X1*X1^T (diag = |x1|^2),
//   C2 = X2*X2^T, Ct1 = X1*[s s ... s] (any col = <x1,s>), Ct2 = X2*[s...].
// f32 16x16x4 A and B per-lane layouts coincide (M/N = lane%16,
// K = 2*(lane/16)+{0,1}) so one float2 load per operand serves both roles.
// EXEC is all-ones for the whole WMMA section (blockDim=32, no divergence).
// ---------------------------------------------------------------------------
__global__ __launch_bounds__(32) void k3_pairs(const float* __restrict__ x,
                                               const float* __restrict__ s,
                                               float* __restrict__ row_loss) {
    const int lane = threadIdx.x;                   // 0..31
    const int i0   = blockIdx.x * 16;
    const int r    = lane & 15;
    const int kh   = (lane >> 4) << 1;              // 0 or 2

    const float* r1 = x + (size_t)(i0 + r) * DD + kh;        // first-half row
    const float* r2 = x + (size_t)(BB + i0 + r) * DD + kh;   // paired row
    const float* sp = s + kh;

    v8f cp  = {}; v8f c1  = {}; v8f c2  = {}; v8f ct1 = {}; v8f ct2 = {};

    for (int k = 0; k < DD; k += 4) {
        v2f a  = *(const v2f*)(r1 + k);
        v2f b  = *(const v2f*)(r2 + k);
        v2f bs = *(const v2f*)(sp + k);
        cp  = __builtin_amdgcn_wmma_f32_16x16x4_f32(false, a, false, b,  (short)0, cp,  false, false);
        c1  = __builtin_amdgcn_wmma_f32_16x16x4_f32(false, a, false, a,  (short)0, c1,  false, false);
        c2  = __builtin_amdgcn_wmma_f32_16x16x4_f32(false, b, false, b,  (short)0, c2,  false, false);
        ct1 = __builtin_amdgcn_wmma_f32_16x16x4_f32(false, a, false, bs, (short)0, ct1, false, false);
        ct2 = __builtin_amdgcn_wmma_f32_16x16x4_f32(false, b, false, bs, (short)0, ct2, false, false);
    }

    // Scatter accumulators to LDS so lane m can read diagonal element (m,m).
    // VGPR v at lane L holds C[M][N], M = v + 8*(L>>4), N = L&15.
    __shared__ float dp[256], d1[256], d2[256], dt1[256], dt2[256];
    const int mhi = (lane >> 4) * 8;
#pragma unroll
    for (int v = 0; v < 8; ++v) {
        const int idx = (v + mhi) * 16 + r;
        dp[idx]  = cp[v];
        d1[idx]  = c1[v];
        d2[idx]  = c2[v];
        dt1[idx] = ct1[v];
        dt2[idx] = ct2[v];
    }
    __syncthreads();

    if (lane < 16) {
        const int m  = lane;
        const int dg = m * 16 + m;
        const float p   = dp[dg];
        const float n1  = fmaxf(sqrtf(d1[dg]), EPSR);
        const float n2  = fmaxf(sqrtf(d2[dg]), EPSR);
        const float inv = 1.0f / TEMPR;
        const float pos  = p / (n1 * n2) * inv;
        const float neg1 = dt1[dg] / n1 * inv;
        const float neg2 = dt2[dg] / n2 * inv;
        auto lse = [](float aa, float bb) {
            float mx = fmaxf(aa, bb);
            return mx + logf(expf(aa - mx) + expf(bb - mx));
        };
        // loss contribution of rows i and i+B: (lse - pos) each
        row_loss[i0 + m] = (lse(pos, neg1) - pos) + (lse(pos, neg2) - pos);
    }
}

// ---------------------------------------------------------------------------
// Kernel 4: mean over 2B rows (row_loss holds per-pair sums of two rows).
// ---------------------------------------------------------------------------
__global__ __launch_bounds__(256) void k4_final(const float* __restrict__ row_loss,
                                                float* __restrict__ out) {
    __shared__ float red[256];
    float acc = 0.f;
    for (int i = threadIdx.x; i < BB; i += 256) acc += row_loss[i];
    red[threadIdx.x] = acc;
    __syncthreads();
    for (int st = 128; st > 0; st >>= 1) {
        if (threadIdx.x < st) red[threadIdx.x] += red[threadIdx.x + st];
        __syncthreads();
    }
    if (threadIdx.x == 0) out[0] = red[0] * (1.0f / (2.0f * BB));
}

extern "C" void kernel_launch(void* const* d_in, const int* in_sizes, int n_in,
                              void* d_out, int out_size, void* d_ws, size_t ws_size,
                              hipStream_t stream) {
    const float* x   = (const float*)d_in[0];   // [2B, D] image_features
    const float* mem = (const float*)d_in[1];   // [M, D]  memory_bank_features
    float* out = (float*)d_out;                 // [1]
    float* ws  = (float*)d_ws;

    float* partial  = ws;                       // NB1 * D floats (1 MB)
    float* svec     = partial + (size_t)NB1 * DD;  // D floats
    float* row_loss = svec + DD;                // B floats

    k1_mem_partial <<<NB1,      256, 0, stream>>>(mem, partial);
    k2_sum_partials<<<DD / 256, 256, 0, stream>>>(partial, svec);
    k3_pairs       <<<BB / 16,  32,  0, stream>>>(x, svec, row_loss);
    k4_final       <<<1,        256, 0, stream>>>(row_loss, out);
}